// TokenPairComponent_89455578841186
// MI455X (gfx1250) — compile-verified
//
#include <hip/hip_runtime.h>
#include <hip/hip_bf16.h>

typedef __attribute__((ext_vector_type(2))) float v2f;
typedef __attribute__((ext_vector_type(8))) float v8f;

#define WMMA_F32(a, b, c) \
    __builtin_amdgcn_wmma_f32_16x16x4_f32(false, (a), false, (b), (short)0, (c), false, false)

// Problem constants (fixed by the reference)
#define BB   4
#define LL   512
#define HH   768
#define HSZ  64
#define OUTN 16
#define PN   131328   // L*(L+1)/2
#define NT   32       // L/16

// ---------------------------------------------------------------------------
// Stage 1: q = x@Wq + bq, k = x@Wk + bk.  One wave computes a 16x64 tile of
// both q and k (shared A fragment, 8 f32 accumulators, K=768 in steps of 4).
// A-fragment loads are contiguous -> single global_load_b64 per step.
// Grid: 128 blocks (B*L/16 row tiles) x 32 threads.
// ---------------------------------------------------------------------------
__global__ void __launch_bounds__(32)
proj_kernel(const float* __restrict__ x, const float* __restrict__ Wq,
            const float* __restrict__ bq, const float* __restrict__ Wk,
            const float* __restrict__ bk,
            float* __restrict__ q, float* __restrict__ k)
{
    const int tile  = blockIdx.x;            // row tile over B*L
    const int lane  = threadIdx.x;
    const int rn    = lane & 15;             // A row / B col within tile
    const int khalf = (lane >> 4) * 2;       // K split across lane halves

    const float* xrow = x + (size_t)(tile * 16 + rn) * HH;

    v8f accq[4] = {{0.f}, {0.f}, {0.f}, {0.f}};
    v8f acck[4] = {{0.f}, {0.f}, {0.f}, {0.f}};

    for (int kk = 0; kk < HH; kk += 4) {
        const int k0 = kk + khalf;
        const v2f a = *(const v2f*)(xrow + k0);   // 8B-aligned contiguous pair
#pragma unroll
        for (int nt = 0; nt < 4; ++nt) {
            const int n = nt * 16 + rn;
            v2f bq2, bk2;                          // 256B row stride: stays b32
            bq2.x = Wq[(size_t)k0 * HSZ + n];
            bq2.y = Wq[(size_t)(k0 + 1) * HSZ + n];
            bk2.x = Wk[(size_t)k0 * HSZ + n];
            bk2.y = Wk[(size_t)(k0 + 1) * HSZ + n];
            accq[nt] = WMMA_F32(a, bq2, accq[nt]);
            acck[nt] = WMMA_F32(a, bk2, acck[nt]);
        }
    }

    const int mbase = (lane >> 4) * 8;
#pragma unroll
    for (int nt = 0; nt < 4; ++nt) {
#pragma unroll
        for (int v = 0; v < 8; ++v) {
            const int gr = tile * 16 + mbase + v;
            const int gc = nt * 16 + rn;
            q[(size_t)gr * HSZ + gc] = accq[nt][v] + bq[gc];
            k[(size_t)gr * HSZ + gc] = acck[nt][v] + bk[gc];
        }
    }
}

// ---------------------------------------------------------------------------
// Stage 2: per-token pair contributions.
//   Acon[i] = q[i] @ Wb[0:64]
//   Ccon[j] = k[j] @ (Wb[64:128]+Wb[192:256]) + q[j] @ Wb[128:192] + bb
// One wave per 16-row tile, N=16, K=64 (16 WMMA steps x 3 GEMMs).
// ---------------------------------------------------------------------------
__global__ void __launch_bounds__(32)
contrib_kernel(const float* __restrict__ q, const float* __restrict__ k,
               const float* __restrict__ Wb, const float* __restrict__ bb,
               float* __restrict__ Acon, float* __restrict__ Ccon)
{
    const int tile  = blockIdx.x;
    const int lane  = threadIdx.x;
    const int rn    = lane & 15;
    const int khalf = (lane >> 4) * 2;

    const float* qrow = q + (size_t)(tile * 16 + rn) * HSZ;
    const float* krow = k + (size_t)(tile * 16 + rn) * HSZ;

    v8f accA = {0.f};
    v8f accC = {0.f};

#pragma unroll
    for (int kk = 0; kk < HSZ; kk += 4) {
        const int k0 = kk + khalf;
        const v2f aq = *(const v2f*)(qrow + k0);   // contiguous b64 loads
        const v2f ak = *(const v2f*)(krow + k0);
        v2f b1, b24, b3;                           // 64B row stride: b32
        b1.x  = Wb[(size_t)k0 * OUTN + rn];
        b1.y  = Wb[(size_t)(k0 + 1) * OUTN + rn];
        b24.x = Wb[(size_t)(64 + k0) * OUTN + rn]     + Wb[(size_t)(192 + k0) * OUTN + rn];
        b24.y = Wb[(size_t)(64 + k0 + 1) * OUTN + rn] + Wb[(size_t)(192 + k0 + 1) * OUTN + rn];
        b3.x  = Wb[(size_t)(128 + k0) * OUTN + rn];
        b3.y  = Wb[(size_t)(128 + k0 + 1) * OUTN + rn];
        accA = WMMA_F32(aq, b1, accA);
        accC = WMMA_F32(ak, b24, accC);
        accC = WMMA_F32(aq, b3, accC);
    }

    const int mbase = (lane >> 4) * 8;
    const float bias = bb[rn];
#pragma unroll
    for (int v = 0; v < 8; ++v) {
        const int gr = tile * 16 + mbase + v;
        Acon[(size_t)gr * OUTN + rn] = accA[v];
        Ccon[(size_t)gr * OUTN + rn] = accC[v] + bias;
    }
}

// ---------------------------------------------------------------------------
// Stage 3: for each upper-triangular 16x16 (s,e) tile: score = q @ k^T via
// WMMA (K=64), then out[b, p(s,e), :] = score + Acon[s] + Ccon[e] for s<=e.
// Both fragments load as contiguous b64; output streamed as 4x b128 stores.
// Grid: (528 tiles, B) x 32 threads.
// ---------------------------------------------------------------------------
__global__ void __launch_bounds__(32)
pair_kernel(const float* __restrict__ q, const float* __restrict__ k,
            const float* __restrict__ Acon, const float* __restrict__ Ccon,
            float* __restrict__ out)
{
    const int b = blockIdx.y;

    // map flat tile id -> (ts, te) with te >= ts
    int t = blockIdx.x;
    int ts = 0, rl = NT;
    while (t >= rl) { t -= rl; --rl; ++ts; }
    const int te = ts + t;

    const int lane  = threadIdx.x;
    const int rn    = lane & 15;
    const int khalf = (lane >> 4) * 2;

    const float* qrow = q + (size_t)(b * LL + ts * 16 + rn) * HSZ; // A rows
    const float* krow = k + (size_t)(b * LL + te * 16 + rn) * HSZ; // B cols (k^T)

    v8f acc = {0.f};
#pragma unroll
    for (int kk = 0; kk < HSZ; kk += 4) {
        const int k0 = kk + khalf;
        const v2f a  = *(const v2f*)(qrow + k0);
        const v2f bv = *(const v2f*)(krow + k0);
        acc = WMMA_F32(a, bv, acc);
    }

    const int e = te * 16 + rn;                 // this lane's end-token column
    const float4* Crow = (const float4*)(Ccon + (size_t)(b * LL + e) * OUTN);
    const float4 c0 = Crow[0], c1 = Crow[1], c2 = Crow[2], c3 = Crow[3];

    const int mbase = (lane >> 4) * 8;
#pragma unroll
    for (int v = 0; v < 8; ++v) {
        const int s = ts * 16 + mbase + v;
        if (s > e) continue;                    // upper triangle only
        const float sc = acc[v];
        const float4* Arow = (const float4*)(Acon + (size_t)(b * LL + s) * OUTN);
        const float4 a0 = Arow[0], a1 = Arow[1], a2 = Arow[2], a3 = Arow[3];
        const int p = s * LL - (s * (s - 1)) / 2 + (e - s);
        float4* o4 = (float4*)(out + ((size_t)b * PN + p) * OUTN);
        o4[0] = make_float4(sc + a0.x + c0.x, sc + a0.y + c0.y, sc + a0.z + c0.z, sc + a0.w + c0.w);
        o4[1] = make_float4(sc + a1.x + c1.x, sc + a1.y + c1.y, sc + a1.z + c1.z, sc + a1.w + c1.w);
        o4[2] = make_float4(sc + a2.x + c2.x, sc + a2.y + c2.y, sc + a2.z + c2.z, sc + a2.w + c2.w);
        o4[3] = make_float4(sc + a3.x + c3.x, sc + a3.y + c3.y, sc + a3.z + c3.z, sc + a3.w + c3.w);
    }
}

extern "C" void kernel_launch(void* const* d_in, const int* in_sizes, int n_in,
                              void* d_out, int out_size, void* d_ws, size_t ws_size,
                              hipStream_t stream) {
    (void)in_sizes; (void)n_in; (void)out_size; (void)ws_size;

    const float* x  = (const float*)d_in[0];
    const float* Wq = (const float*)d_in[1];
    const float* bq = (const float*)d_in[2];
    const float* Wk = (const float*)d_in[3];
    const float* bk = (const float*)d_in[4];
    const float* Wb = (const float*)d_in[5];
    const float* bb = (const float*)d_in[6];
    float* out = (float*)d_out;

    // Workspace layout (floats): q (2048*64), k (2048*64), Acon (2048*16),
    // Ccon (2048*16)  -> ~1.31 MB total, trivially L2-resident.
    float* ws   = (float*)d_ws;
    float* q    = ws;
    float* k    = ws + (size_t)BB * LL * HSZ;
    float* Acon = k  + (size_t)BB * LL * HSZ;
    float* Ccon = Acon + (size_t)BB * LL * OUTN;

    const int rowTiles  = (BB * LL) / 16;      // 128
    const int pairTiles = (NT * (NT + 1)) / 2; // 528

    proj_kernel<<<dim3(rowTiles), dim3(32), 0, stream>>>(x, Wq, bq, Wk, bk, q, k);
    contrib_kernel<<<dim3(rowTiles), dim3(32), 0, stream>>>(q, k, Wb, bb, Acon, Ccon);
    pair_kernel<<<dim3(pairTiles, BB), dim3(32), 0, stream>>>(q, k, Acon, Ccon, out);
}